// BYOL_TT_Loss_30064771072566
// MI455X (gfx1250) — compile-verified
//
#include <hip/hip_runtime.h>
#include <hip/hip_bf16.h>

// BYOL loss: only the diagonal of the cosine-sim matrix is used.
// loss = 2 - 2 * mean_i( dot(x_i, xt_i) / (max(|x_i|,eps)*max(|xt_i|,eps)) )
// O(N*D) memory-bound reduction (32 MB @ 23.3 TB/s ~ 1.4us), fp32 throughout.
// Per-tile Gram diagonals computed with V_WMMA_F32_16X16X4_F32 (fp32 accumulate).

typedef __attribute__((ext_vector_type(2))) float v2f;
typedef __attribute__((ext_vector_type(8))) float v8f;

#define BYOL_D   512
#define BYOL_EPS 1e-8f

// A-fragment for 16x16x4 f32 WMMA from row-major M[. , 512], rows row0..row0+15:
//   lane l (l<16):  row=l,     holds (K=k+0, K=k+1)
//   lane l (>=16):  row=l-16,  holds (K=k+2, K=k+3)
// The SAME per-lane load also forms the B-fragment of M^T (B[k][n] = M[n][k]):
//   VGPR0: lanes0-15 = B[k+0][n], lanes16-31 = B[k+2][n]; VGPR1: k+1 / k+3.
__device__ __forceinline__ v2f byol_load_frag(const float* __restrict__ M,
                                              int row0, int k, int lane) {
  int r  = row0 + (lane & 15);
  int kk = k + ((lane >> 4) << 1);
  return *(const v2f*)(M + (size_t)r * BYOL_D + kk);
}

// C/D layout: VGPR r, lanes 0-15 -> M=r, N=lane; lanes 16-31 -> M=r+8, N=lane-16.
// diag(m): m<8  at (vgpr m,   lane m)
//          m>=8 at (vgpr m-8, lane m+16)
// Returns diag[lane] valid for lanes 0..15.
__device__ __forceinline__ float byol_diag(v8f c, int lane) {
  int idx = (lane < 8) ? lane : (lane - 24);  // meaningful for lanes 0-7 and 24-31
  float v = c[0];
#pragma unroll
  for (int r = 1; r < 8; ++r) v = (idx == r) ? c[r] : v;
  float hi = __shfl(v, (lane + 16) & 31, 32); // lanes 8-15 fetch from lanes 24-31
  return (lane < 8) ? v : hi;
}

__global__ __launch_bounds__(256) void byol_diag_cos_kernel(
    const float* __restrict__ x, const float* __restrict__ xt,
    float* __restrict__ per_row, int n_tiles) {
  const int lane = threadIdx.x & 31;
  const int wave = threadIdx.x >> 5;
  const int tile = blockIdx.x * (blockDim.x >> 5) + wave;
  if (tile >= n_tiles) return;            // wave-uniform: EXEC stays all-ones
  const int row0 = tile << 4;

  v8f acc_d = {};  // diag -> dot(x_i, xt_i)
  v8f acc_x = {};  // diag -> |x_i|^2
  v8f acc_t = {};  // diag -> |xt_i|^2

#pragma unroll 4
  for (int k = 0; k < BYOL_D; k += 4) {
    v2f fa = byol_load_frag(x,  row0, k, lane);
    v2f fb = byol_load_frag(xt, row0, k, lane);
    acc_d = __builtin_amdgcn_wmma_f32_16x16x4_f32(false, fa, false, fb,
                                                  (short)0, acc_d, false, false);
    acc_x = __builtin_amdgcn_wmma_f32_16x16x4_f32(false, fa, false, fa,
                                                  (short)0, acc_x, false, false);
    acc_t = __builtin_amdgcn_wmma_f32_16x16x4_f32(false, fb, false, fb,
                                                  (short)0, acc_t, false, false);
  }

  float dot = byol_diag(acc_d, lane);
  float nx  = byol_diag(acc_x, lane);
  float nt  = byol_diag(acc_t, lane);

  if (lane < 16) {
    float denom = fmaxf(sqrtf(nx), BYOL_EPS) * fmaxf(sqrtf(nt), BYOL_EPS);
    per_row[row0 + lane] = dot / denom;
  }
}

__global__ __launch_bounds__(256) void byol_reduce_kernel(
    const float* __restrict__ per_row, float* __restrict__ out, int n) {
  __shared__ float sm[256];
  float s = 0.f;
  for (int i = threadIdx.x; i < n; i += 256) s += per_row[i];
  sm[threadIdx.x] = s;
  __syncthreads();
#pragma unroll
  for (int off = 128; off > 0; off >>= 1) {
    if ((int)threadIdx.x < off) sm[threadIdx.x] += sm[threadIdx.x + off];
    __syncthreads();
  }
  if (threadIdx.x == 0) out[0] = 2.f - 2.f * (sm[0] / (float)n);
}

extern "C" void kernel_launch(void* const* d_in, const int* in_sizes, int n_in,
                              void* d_out, int out_size, void* d_ws, size_t ws_size,
                              hipStream_t stream) {
  const float* x  = (const float*)d_in[0];
  const float* xt = (const float*)d_in[1];
  float* out      = (float*)d_out;
  float* per_row  = (float*)d_ws;        // N floats of scratch

  const int N       = in_sizes[0] / BYOL_D;   // 8192
  const int n_tiles = N >> 4;                 // 512 tiles of 16 rows
  const int waves_per_block = 256 / 32;       // 8
  const int blocks = (n_tiles + waves_per_block - 1) / waves_per_block;

  byol_diag_cos_kernel<<<blocks, 256, 0, stream>>>(x, xt, per_row, n_tiles);
  byol_reduce_kernel<<<1, 256, 0, stream>>>(per_row, out, N);
}